// QINRLayer_55490977464500
// MI455X (gfx1250) — compile-verified
//
#include <hip/hip_runtime.h>
#include <math.h>

#define BATCH 8192
#define FEAT  40
#define NW    10
#define DIM   1024
#define NBLK  4
#define BN_EPS 1e-5f

typedef __attribute__((ext_vector_type(2))) float v2f;
typedef __attribute__((ext_vector_type(8))) float v8f;

// ============================================================================
// WMMA GEMM:  out[m, coloff+n] = act( X[M,K] @ W[N,K]^T + bias[n] )
// One wave per 16x16 output tile, V_WMMA_F32_16X16X4_F32, K stepped by 4.
// A frag layout (16x4 f32): lanes 0-15 hold K=0,1 ; lanes 16-31 hold K=2,3.
// C/D layout: VGPR r -> M = r (lanes 0-15) / r+8 (lanes 16-31), N = lane&15.
// Out-of-range B columns: clamp the address (their D columns are never
// stored), so the K-loop has no divergence -> EXEC stays all-1s for WMMA.
// ============================================================================
__global__ __launch_bounds__(256)
void gemm_wmma_f32(const float* __restrict__ X, const float* __restrict__ W,
                   const float* __restrict__ bias, float* __restrict__ out,
                   int M, int N, int K, int ldout, int coloff, int relu)
{
    int gtid = blockIdx.x * blockDim.x + threadIdx.x;
    int wave = gtid >> 5;
    int lane = gtid & 31;
    int ntiles = (N + 15) >> 4;
    int mtiles = M >> 4;
    if (wave >= mtiles * ntiles) return;   // grid sized exactly; never divergent
    int mt = wave / ntiles, nt = wave % ntiles;
    int m0 = mt << 4, n0 = nt << 4;
    int hb  = lane >> 4;     // wave half selects K pair {0,1} vs {2,3}
    int l16 = lane & 15;
    int row  = m0 + l16;                     // < M always (M multiple of 16)
    int col  = n0 + l16;                     // may exceed N
    int colc = (col < N) ? col : (N - 1);    // clamped address, no branch

    const float2* Xp = (const float2*)(X + row * K);    // K even, ka even
    const float2* Wp = (const float2*)(W + colc * K);

    v8f acc = {};
    for (int k0 = 0; k0 < K; k0 += 4) {
        int kh = (k0 >> 1) + hb;             // float2 index of K pair
        float2 av = Xp[kh];
        float2 bv = Wp[kh];
        v2f a, b;
        a[0] = av.x; a[1] = av.y;
        b[0] = bv.x; b[1] = bv.y;
        acc = __builtin_amdgcn_wmma_f32_16x16x4_f32(
            /*neg_a=*/false, a, /*neg_b=*/false, b,
            /*c_mod=*/(short)0, acc, /*reuse_a=*/false, /*reuse_b=*/false);
    }
    if (col < N) {
        float bvadd = bias[col];
#pragma unroll
        for (int r = 0; r < 8; r++) {
            int m = m0 + r + hb * 8;
            float v = acc[r] + bvadd;
            if (relu) v = v > 0.f ? v : 0.f;
            out[m * ldout + coloff + col] = v;
        }
    }
}

// ============================================================================
// BatchNorm statistics: per-feature scale/shift so QNN can fuse normalization
// as angle = xq*scale + shift.
// ============================================================================
__global__ __launch_bounds__(256)
void bn_stats(const float* __restrict__ xq, const float* __restrict__ gamma,
              const float* __restrict__ beta, float* __restrict__ scale,
              float* __restrict__ shift)
{
    __shared__ float s1[256], s2[256];
    int f = blockIdx.x;
    int t = threadIdx.x;
    float a = 0.f, b = 0.f;
    for (int r = t; r < BATCH; r += 256) {
        float v = xq[r * FEAT + f];
        a += v; b += v * v;
    }
    s1[t] = a; s2[t] = b;
    __syncthreads();
    for (int o = 128; o; o >>= 1) {
        if (t < o) { s1[t] += s1[t + o]; s2[t] += s2[t + o]; }
        __syncthreads();
    }
    if (t == 0) {
        float mean = s1[0] * (1.f / BATCH);
        float var  = s2[0] * (1.f / BATCH) - mean * mean;
        float sc   = gamma[f] * rsqrtf(var + BN_EPS);
        scale[f] = sc;
        shift[f] = beta[f] - mean * sc;
    }
}

// ============================================================================
// QNN: one wave32 per sample, 1024-amplitude complex state in an 8KB LDS
// slice owned exclusively by the wave -> zero barriers (per-wave DS ordering).
// wire w acts on bit position p = 9 - w (wire 0 = MSB of flattened index).
// ============================================================================
__device__ __forceinline__ void apply_real(float2* st, int p,
                                           float g00, float g01, float g10, float g11,
                                           int lane)
{
    int mask = 1 << p;
    for (int j = lane; j < (DIM >> 1); j += 32) {
        int lo = j & (mask - 1);
        int i0 = ((j ^ lo) << 1) | lo;
        int i1 = i0 | mask;
        float2 a = st[i0], b = st[i1];
        st[i0] = make_float2(g00 * a.x + g01 * b.x, g00 * a.y + g01 * b.y);
        st[i1] = make_float2(g10 * a.x + g11 * b.x, g10 * a.y + g11 * b.y);
    }
}

__device__ __forceinline__ void apply_rx(float2* st, int p, float c, float s, int lane)
{   // [[c, -is],[-is, c]]
    int mask = 1 << p;
    for (int j = lane; j < (DIM >> 1); j += 32) {
        int lo = j & (mask - 1);
        int i0 = ((j ^ lo) << 1) | lo;
        int i1 = i0 | mask;
        float2 a = st[i0], b = st[i1];
        st[i0] = make_float2(c * a.x + s * b.y, c * a.y - s * b.x);
        st[i1] = make_float2(c * b.x + s * a.y, c * b.y - s * a.x);
    }
}

__device__ __forceinline__ void apply_rz(float2* st, int p, float c, float s, int lane)
{   // diag(c - is, c + is)
    int mask = 1 << p;
    for (int k = lane; k < DIM; k += 32) {
        float2 v = st[k];
        float ss = (k & mask) ? s : -s;
        st[k] = make_float2(c * v.x - ss * v.y, c * v.y + ss * v.x);
    }
}

__device__ __forceinline__ void apply_gen(float2* st, int p,
                                          float2 g00, float2 g01, float2 g10, float2 g11,
                                          int lane)
{
    int mask = 1 << p;
    for (int j = lane; j < (DIM >> 1); j += 32) {
        int lo = j & (mask - 1);
        int i0 = ((j ^ lo) << 1) | lo;
        int i1 = i0 | mask;
        float2 a = st[i0], b = st[i1];
        st[i0] = make_float2(g00.x * a.x - g00.y * a.y + g01.x * b.x - g01.y * b.y,
                             g00.x * a.y + g00.y * a.x + g01.x * b.y + g01.y * b.x);
        st[i1] = make_float2(g10.x * a.x - g10.y * a.y + g11.x * b.x - g11.y * b.y,
                             g10.x * a.y + g10.y * a.x + g11.x * b.y + g11.y * b.x);
    }
}

__device__ __forceinline__ void apply_cnot(float2* st, int pc, int pt, int lane)
{
    int plo = pc < pt ? pc : pt;
    int phi = pc < pt ? pt : pc;
    for (int j = lane; j < (DIM >> 2); j += 32) {
        int lo  = j & ((1 << plo) - 1);
        int t1  = ((j ^ lo) << 1) | lo;          // insert 0 at plo
        int lo2 = t1 & ((1 << phi) - 1);
        int t2  = ((t1 ^ lo2) << 1) | lo2;       // insert 0 at phi
        int i0  = t2 | (1 << pc);                // control=1, target=0
        int i1  = i0 | (1 << pt);
        float2 tmp = st[i0]; st[i0] = st[i1]; st[i1] = tmp;
    }
}

__device__ __forceinline__ void measure_write(const float2* st, int lane, float* dst)
{
    float acc[NW];
#pragma unroll
    for (int w = 0; w < NW; w++) acc[w] = 0.f;
    for (int k = lane; k < DIM; k += 32) {
        float2 v = st[k];
        float pr = v.x * v.x + v.y * v.y;
#pragma unroll
        for (int w = 0; w < NW; w++)
            acc[w] += ((k >> (NW - 1 - w)) & 1) ? -pr : pr;
    }
#pragma unroll
    for (int w = 0; w < NW; w++)
        for (int o = 16; o; o >>= 1)
            acc[w] += __shfl_xor(acc[w], o, 32);
    if (lane == 0) {
#pragma unroll
        for (int w = 0; w < NW; w++) dst[w] = acc[w];
    }
}

__global__ __launch_bounds__(256)
void qnn_kernel(const float* __restrict__ xq, const float* __restrict__ scale,
                const float* __restrict__ shift,
                const float* __restrict__ rz1, const float* __restrict__ ry1,
                const float* __restrict__ rz2, const float* __restrict__ rz3,
                float* __restrict__ xcat)
{
    __shared__ float2 S[8 * DIM];                  // 64 KB: 8 samples/block
    int wv = threadIdx.x >> 5, lane = threadIdx.x & 31;
    int b = blockIdx.x * 8 + wv;
    float2* st = &S[wv * DIM];

    for (int k = lane; k < DIM; k += 32)
        st[k] = make_float2(k == 0 ? 1.f : 0.f, 0.f);

    const float* xrow = xq + b * FEAT;

    for (int kb = 0; kb < NBLK; kb++) {
        for (int i = 0; i < NW; i++) {
            float th = 0.5f * rz1[kb * NW + i];
            apply_rz(st, NW - 1 - i, __cosf(th), __sinf(th), lane);
        }
        for (int i = 0; i < NW; i++) {
            float th = 0.5f * ry1[kb * NW + i];
            float c = __cosf(th), s = __sinf(th);
            apply_real(st, NW - 1 - i, c, -s, s, c, lane);
        }
        for (int i = 0; i < NW; i++) {
            float th = 0.5f * rz2[kb * NW + i];
            apply_rz(st, NW - 1 - i, __cosf(th), __sinf(th), lane);
        }
        for (int i = 0; i < NW; i++)
            apply_cnot(st, NW - 1 - i, NW - 1 - ((i + 1) % NW), lane);
        if (kb != NBLK - 1) {
            for (int i = 0; i < NW; i++) {
                int p = NW - 1 - i;
                float a0 = 0.5f * (xrow[4*i+0] * scale[4*i+0] + shift[4*i+0]);
                float a1 = 0.5f * (xrow[4*i+1] * scale[4*i+1] + shift[4*i+1]);
                float a2 = 0.5f * (xrow[4*i+2] * scale[4*i+2] + shift[4*i+2]);
                float a3 = 0.5f * (xrow[4*i+3] * scale[4*i+3] + shift[4*i+3]);
                apply_rx(st, p, __cosf(a0), __sinf(a0), lane);
                {   float c = __cosf(a1), s = __sinf(a1);
                    apply_real(st, p, c, -s, s, c, lane); }
                apply_rz(st, p, __cosf(a2), __sinf(a2), lane);
                apply_rx(st, p, __cosf(a3), __sinf(a3), lane);
            }
        }
    }

    float* orow = xcat + b * (2 * FEAT);
    measure_write(st, lane, orow);                          // cols 0..9

    const float R = 0.70710678f;
    for (int i = 0; i < NW; i++)
        apply_real(st, NW - 1 - i, R, R, R, -R, lane);      // H on all wires
    measure_write(st, lane, orow + NW);                     // cols 10..19

    for (int i = 0; i < NW; i++) {                          // H then SX per wire
        int p = NW - 1 - i;
        apply_real(st, p, R, R, R, -R, lane);
        float2 g00 = make_float2(0.5f, 0.5f);
        float2 g01 = make_float2(0.5f, -0.5f);
        apply_gen(st, p, g00, g01, g01, g00, lane);
    }
    measure_write(st, lane, orow + 2 * NW);                 // cols 20..29

    for (int i = 0; i < NW; i++) {
        float th = 0.5f * rz3[(NBLK - 1) * NW + i];
        apply_rz(st, NW - 1 - i, __cosf(th), __sinf(th), lane);
    }
    measure_write(st, lane, orow + 3 * NW);                 // cols 30..39
}

// ============================================================================
extern "C" void kernel_launch(void* const* d_in, const int* in_sizes, int n_in,
                              void* d_out, int out_size, void* d_ws, size_t ws_size,
                              hipStream_t stream)
{
    (void)in_sizes; (void)n_in; (void)out_size; (void)ws_size;
    const float* x        = (const float*)d_in[0];
    const float* qlin_w   = (const float*)d_in[1];
    const float* qlin_b   = (const float*)d_in[2];
    const float* bn_gamma = (const float*)d_in[3];
    const float* bn_beta  = (const float*)d_in[4];
    const float* clin_w   = (const float*)d_in[5];
    const float* clin_b   = (const float*)d_in[6];
    const float* lin_w    = (const float*)d_in[7];
    const float* lin_b    = (const float*)d_in[8];
    const float* rz1      = (const float*)d_in[9];
    const float* ry1      = (const float*)d_in[10];
    const float* rz2      = (const float*)d_in[11];
    const float* rz3      = (const float*)d_in[12];
    float* ws   = (float*)d_ws;
    float* xq   = ws;                               // [8192, 40]
    float* xcat = ws + BATCH * FEAT;                // [8192, 80] = [x1 | x2]
    float* sc   = xcat + BATCH * 2 * FEAT;          // [40]
    float* sh   = sc + FEAT;                        // [40]
    float* out  = (float*)d_out;                    // [8192, 40]

    int mt = BATCH / 16, nt = (FEAT + 15) / 16;
    int blocks = (mt * nt * 32 + 255) / 256;

    // quantum-branch pre-linear and classical branch (relu into xcat[:,40:80])
    gemm_wmma_f32<<<blocks, 256, 0, stream>>>(x, qlin_w, qlin_b, xq,
                                              BATCH, FEAT, FEAT, FEAT, 0, 0);
    gemm_wmma_f32<<<blocks, 256, 0, stream>>>(x, clin_w, clin_b, xcat,
                                              BATCH, FEAT, FEAT, 2 * FEAT, FEAT, 1);
    bn_stats<<<FEAT, 256, 0, stream>>>(xq, bn_gamma, bn_beta, sc, sh);
    qnn_kernel<<<BATCH / 8, 256, 0, stream>>>(xq, sc, sh, rz1, ry1, rz2, rz3, xcat);
    // final linear over concat (K = 80)
    gemm_wmma_f32<<<blocks, 256, 0, stream>>>(xcat, lin_w, lin_b, out,
                                              BATCH, FEAT, 2 * FEAT, FEAT, 0, 0);
}